// Scatter_Rendering_87101936763449
// MI455X (gfx1250) — compile-verified
//
#include <hip/hip_runtime.h>
#include <stdint.h>

#define LENS 11
#define PADR 5
#define TILE 16
#define NTHREADS (TILE*TILE)   // 256
#define HT   (TILE + 2*PADR)   // 26: tile + halo
#define TA   (HT*HT)           // 676 halo-tile pixels
#define HH   512
#define WW   512
#define HWC  (HH*WW)           // per-channel plane stride

typedef __attribute__((address_space(1))) int* as1_i32;  // global ("__device__ int*")
typedef __attribute__((address_space(3))) int* as3_i32;  // LDS    ("__shared__ int*")

// Async global->LDS copy (CDNA5 GLOBAL_LOAD_ASYNC_TO_LDS_B32, ASYNCcnt).
__device__ __forceinline__ void async_cp_b32(void* lds_dst, const float* gsrc) {
#if __has_builtin(__builtin_amdgcn_global_load_async_to_lds_b32)
  __builtin_amdgcn_global_load_async_to_lds_b32(
      (as1_i32)(uint64_t)(uintptr_t)gsrc,   // flat==global aperture
      (as3_i32)lds_dst,                     // generic->LDS addrspacecast
      0, 0);
#else
  *(volatile float*)lds_dst = *gsrc;        // synchronous fallback
#endif
}

__device__ __forceinline__ void async_wait0() {
#if __has_builtin(__builtin_amdgcn_s_wait_asynccnt)
  __builtin_amdgcn_s_wait_asynccnt(0);
#else
  asm volatile("s_wait_asynccnt 0" ::: "memory");
#endif
}

__global__ __launch_bounds__(NTHREADS)
void Scatter_Rendering_87101936763449_kernel(
    const float* __restrict__ x,          // (4,4,512,512)
    const float* __restrict__ lens_eff,   // (4,1)
    const float* __restrict__ diskernel,  // (11,11)
    const float* __restrict__ lens_mask,  // (11,11)
    float* __restrict__ out)              // (4,3,512,512)
{
  // AoS {R,G,B, gweight}: one ds_load_b128 per tap; r in its own plane.
  __shared__ float4 sP[TA];
  __shared__ float  sR[TA];            // disp on load, radius after transform
  __shared__ float  sT[LENS*LENS];     // dist with mask folded in (+INF if masked)

  const int tx  = threadIdx.x, ty = threadIdx.y;
  const int tid = ty * TILE + tx;
  const int b   = blockIdx.z;
  const int x0  = blockIdx.x * TILE;
  const int y0  = blockIdx.y * TILE;

  // Fold lens_mask into the distance table: cover = (r >= t) implies mask.
  if (tid < LENS * LENS) {
    float m  = lens_mask[tid];
    sT[tid]  = (m != 0.0f) ? diskernel[tid] : 3.0e38f;
  }

  const float  lens = lens_eff[b];                 // uniform per block -> SGPR
  const float* xb   = x + (size_t)b * 4 * HWC;

  // Stage 1: async-copy the 26x26 halo tile (4 channels) straight into LDS.
  // 676 = 2*256 + 164 -> two unguarded rounds + one guarded round, so the
  // async copies issue as one batch per thread with minimal control flow.
  #pragma unroll
  for (int it = 0; it < 3; ++it) {
    const int p = tid + it * NTHREADS;
    if (it < 2 || p < TA) {
      int hy = p / HT, hx = p - hy * HT;
      int gy = y0 + hy - PADR; gy = gy < 0 ? 0 : (gy > HH - 1 ? HH - 1 : gy);
      int gx = x0 + hx - PADR; gx = gx < 0 ? 0 : (gx > WW - 1 ? WW - 1 : gx);
      const float* src = xb + (size_t)gy * WW + gx;
      float* dst = (float*)&sP[p];
      async_cp_b32(dst + 0, src);              // R
      async_cp_b32(dst + 1, src + HWC);        // G
      async_cp_b32(dst + 2, src + 2 * HWC);    // B
      async_cp_b32(&sR[p],  src + 3 * HWC);    // disp
    }
  }
  async_wait0();        // drain this wave's ASYNCcnt before the barrier
  __syncthreads();

  // Stage 2: disp -> (g, r), once per tile pixel (amortizes the reciprocal
  // over all 121 taps that read this pixel). Each pixel owned by one thread.
  #pragma unroll
  for (int it = 0; it < 3; ++it) {
    const int p = tid + it * NTHREADS;
    if (it < 2 || p < TA) {
      float d = sR[p];
      float r = fabsf(d) * lens;
      r = r < (float)PADR ? r : (float)PADR;   // clip to lens//2
      sP[p].w = 1.0f / (3.14159265358979323846f * r * r + 1.0f);
      sR[p]   = r;
    }
  }
  __syncthreads();

  // Stage 3: 121-tap gated accumulation out of LDS.
  // Outer row loop rolled (low pressure); inner 11 taps fully unrolled so the
  // compiler vectorizes the consecutive sR/sT row reads into b128/b96 chunks.
  float n0 = 0.f, n1 = 0.f, n2 = 0.f, den = 0.f;
  #pragma unroll 1
  for (int i = 0; i < LENS; ++i) {
    const int rowq = (ty + i) * HT + tx;
    const int rowk = i * LENS;
    #pragma unroll
    for (int j = 0; j < LENS; ++j) {
      const int q = rowq + j;
      float  t = sT[rowk + j];               // broadcast, row-vectorized
      float4 v = sP[q];                      // ds_load_b128 (rgb + g)
      float  r = sR[q];                      // row-vectorized b32
      float wg = (r >= t) ? v.w : 0.0f;      // cover & mask & g in one select
      n0  = fmaf(wg, v.x, n0);
      n1  = fmaf(wg, v.y, n1);
      n2  = fmaf(wg, v.z, n2);
      den += wg;
    }
  }

  // den > 0 always: the center tap has dist 0, mask 1, so wg >= 1/(25*pi+1).
  const float inv = 1.0f / den;
  size_t o = (size_t)b * 3 * HWC + (size_t)(y0 + ty) * WW + (x0 + tx);
  out[o]           = n0 * inv;
  out[o + HWC]     = n1 * inv;
  out[o + 2 * HWC] = n2 * inv;
}

extern "C" void kernel_launch(void* const* d_in, const int* in_sizes, int n_in,
                              void* d_out, int out_size, void* d_ws, size_t ws_size,
                              hipStream_t stream) {
  (void)in_sizes; (void)n_in; (void)out_size; (void)d_ws; (void)ws_size;
  const float* x    = (const float*)d_in[0];
  const float* lens = (const float*)d_in[1];
  const float* disk = (const float*)d_in[2];
  const float* mask = (const float*)d_in[3];
  float* out = (float*)d_out;

  dim3 block(TILE, TILE, 1);
  dim3 grid(WW / TILE, HH / TILE, 4);
  hipLaunchKernelGGL(Scatter_Rendering_87101936763449_kernel,
                     grid, block, 0, stream, x, lens, disk, mask, out);
}